// ResidualBlock_9113920602599
// MI455X (gfx1250) — compile-verified
//
#include <hip/hip_runtime.h>
#include <math.h>
#include <stdint.h>

#define B_   16
#define CIN  64
#define COUT 128
#define NP   4096            // 64*64 pixels
#define TP   8               // pixels per workgroup == waves per workgroup
#define NCH  (B_*COUT)       // 2048 instance-norm channels
#define EPSN 1e-5f

#define USE_ASYNC_STAGE 1    // async global->LDS staging (CDNA5 ASYNCcnt path)

typedef float v2f __attribute__((ext_vector_type(2)));
typedef float v8f __attribute__((ext_vector_type(8)));

// ---- CDNA5 fp32 WMMA: D(16x16,f32) = A(16x4,f32) * B(4x16,f32) + C ----
#if defined(__gfx1250__) && __has_builtin(__builtin_amdgcn_wmma_f32_16x16x4_f32)
#define WMMA_F32(a, b, c) \
  __builtin_amdgcn_wmma_f32_16x16x4_f32(false, (a), false, (b), (short)0, (c), false, false)
#else
#warning "wmma f32_16x16x4 builtin unavailable on this pass - placeholder fallback in use"
__device__ inline v8f wmma_fallback(v2f a, v2f b, v8f c) { c[0] += a[0]*b[0] + a[1]*b[1]; return c; }
#define WMMA_F32(a, b, c) wmma_fallback((a), (b), (c))
#endif

#if defined(__gfx1250__) && USE_ASYNC_STAGE
// per-lane LDS destination (byte offset), per-lane 32-bit global byte offset + SGPR64 base.
// f1 is a touch-once 256 MB stream -> non-temporal hint keeps L2 for x / output tiles.
#define ASYNC_LD_B64_NT(ldsoff, gvoff, base) \
  asm volatile("global_load_async_to_lds_b64 %0, %1, %2 th:TH_LOAD_NT" \
               :: "v"(ldsoff), "v"(gvoff), "s"(base) : "memory")
#define ASYNC_WAIT0() asm volatile("s_wait_asynccnt 0x0" ::: "memory")
#else
#define ASYNC_LD_B64_NT(ldsoff, gvoff, base) do { (void)(ldsoff); (void)(gvoff); (void)(base); } while (0)
#define ASYNC_WAIT0() do {} while (0)
#endif

__device__ __forceinline__ v2f mk2(float a, float b) { v2f v; v[0] = a; v[1] = b; return v; }
__device__ __forceinline__ float2 cmul(float2 a, float2 b) {
  return make_float2(a.x*b.x - a.y*b.y, a.x*b.y + a.y*b.x);
}
__device__ __forceinline__ float2 cadd(float2 a, float2 b) { return make_float2(a.x+b.x, a.y+b.y); }
__device__ __forceinline__ float2 logmag(float2 z) {
  float m = sqrtf(z.x*z.x + z.y*z.y);
  float s = (m > 0.f) ? (log1pf(m) / m) : 0.f;   // log1p(|z|) * z/|z|, 0 at origin
  return make_float2(z.x*s, z.y*s);
}

// =====================================================================
// Kernel 1: per-pixel complex GEMMs via WMMA, async double-buffered f1 stream.
//   conv1: out1[b, o, p] = sum_i x[b,i,p] * f1[o,i,p]   (per-pixel weights, 256 MB stream)
//   skip : skip[b, o, p] = sum_i x[b,i,p] * fs[o,i]     (shared weights, L0-resident)
// Wave w owns pixel p0+w: M=batch(16), K=Cin(64), N=16-wide Cout tile.
// Staging unit = "chunk" c = 2*ot + kh: Cout-tile ot, K-half kh (i in [32*kh, 32*kh+32)).
// Chunk c lives in LDS buffer kh; chunk c+1 streams in (ASYNCcnt) while c is consumed.
// kh is an unrolled compile-time index so A-fragments stay directly register-addressed
// (no M0/v_movrels indexed-VGPR access in the hot loop).
// =====================================================================
__global__ __launch_bounds__(32*TP) void conv_gemm_kernel(
    const float*  __restrict__ xr, const float* __restrict__ xi,
    const float2* __restrict__ f1, const float2* __restrict__ fsw,
    float2* __restrict__ out1, float2* __restrict__ skipb)
{
  __shared__ float2 xs [CIN][16][TP+1];        // [i][b][p]               73728 B
  __shared__ float2 f1s[2][CIN/2][16][TP+1];   // [buf][i'][o_local][p]   73728 B

  const int tid  = threadIdx.x;
  const int lane = tid & 31;
  const int wave = tid >> 5;        // pixel within tile
  const int mn   = lane & 15;       // A: row m (batch) / B: col n (cout)
  const int hi   = lane >> 4;       // upper-half-lane K offset
  const int p0   = blockIdx.x * TP;
  const int pp   = tid & (TP-1);
  const int r0   = tid / TP;        // 0..31

  // ---- stage x tile synchronously (one-time, 32 MB total across grid) ----
  for (int r = r0; r < B_*CIN; r += 32) {      // r = b*CIN + i
    const int b = r >> 6, i = r & 63;
    const int g = r * NP + p0 + pp;
    xs[i][b][pp] = make_float2(xr[g], xi[g]);
  }

  // ---- async stage of f1 chunk c into buffer c&1 (16 reqs/thread, in-order done) ----
  auto stage_chunk = [&](int c) {
    const int o0 = (c >> 1) * 16;
    const int ib = (c & 1) * 32;
    const int bb = c & 1;
#if defined(__gfx1250__) && USE_ASYNC_STAGE
    for (int r = r0; r < 16*32; r += 32) {     // r = ol*32 + il
      const int ol = r >> 5, il = r & 31;
      const unsigned ldsoff = (unsigned)(uintptr_t)&f1s[bb][il][ol][pp];
      const unsigned gvoff  = (unsigned)((((o0 + ol)*CIN + ib + il)*NP + p0 + pp) * 8u);
      ASYNC_LD_B64_NT(ldsoff, gvoff, f1);
    }
#else
    for (int r = r0; r < 16*32; r += 32) {
      const int ol = r >> 5, il = r & 31;
      f1s[bb][il][ol][pp] = f1[((size_t)(o0 + ol)*CIN + ib + il)*NP + p0 + pp];
    }
#endif
  };

  stage_chunk(0);
  ASYNC_WAIT0();
  __syncthreads();

  // ---- A fragments (x, re/im split): 16 k-steps, held in registers ----
  v2f ar[16], ai[16];
  #pragma unroll
  for (int k = 0; k < 16; ++k) {
    const int kk = 4*k + 2*hi;
    float2 c0 = xs[kk    ][mn][wave];
    float2 c1 = xs[kk + 1][mn][wave];
    ar[k] = mk2(c0.x, c1.x);
    ai[k] = mk2(c0.y, c1.y);
  }

  for (int ot = 0; ot < COUT/16; ++ot) {
    const int o0 = ot * 16;
    v8f RR = {}, II = {}, RI = {}, IR = {};          // conv1 accumulators
    v8f sRR = {}, sII = {}, sRI = {}, sIR = {};      // skip accumulators

    #pragma unroll
    for (int kh = 0; kh < 2; ++kh) {                 // compile-time K-half / LDS buffer
      const int c = 2*ot + kh;
      if (c + 1 < 16) stage_chunk(c + 1);            // overlap: next chunk streams in

      #pragma unroll
      for (int k = 0; k < 8; ++k) {
        const int kl = 4*k + 2*hi;                   // chunk-local i index
        const int kg = kh*8 + k;                     // constant A-fragment index
        // conv1 B fragment from LDS chunk (buffer index kh is compile-time)
        float2 c0 = f1s[kh][kl    ][mn][wave];
        float2 c1 = f1s[kh][kl + 1][mn][wave];
        v2f br = mk2(c0.x, c1.x);
        v2f bi = mk2(c0.y, c1.y);
        RR = WMMA_F32(ar[kg], br, RR);
        II = WMMA_F32(ai[kg], bi, II);
        RI = WMMA_F32(ar[kg], bi, RI);
        IR = WMMA_F32(ai[kg], br, IR);
        // skip B fragment straight from tiny cached fs[o, i]
        const int ig = 4*kg + 2*hi;
        float2 s0 = fsw[(size_t)(o0 + mn)*CIN + ig    ];
        float2 s1 = fsw[(size_t)(o0 + mn)*CIN + ig + 1];
        v2f tr = mk2(s0.x, s1.x);
        v2f ti = mk2(s0.y, s1.y);
        sRR = WMMA_F32(ar[kg], tr, sRR);
        sII = WMMA_F32(ai[kg], ti, sII);
        sRI = WMMA_F32(ar[kg], ti, sRI);
        sIR = WMMA_F32(ai[kg], tr, sIR);
      }

      if (c != 15) {            // uniform condition; last chunk needs no wait/barrier
        ASYNC_WAIT0();          // my async writes for chunk c+1 landed in LDS
        __syncthreads();        // everyone's landed; all readers of chunk c are done
      }
    }

    // ---- Cout tile complete: combine complex parts + store ----
    {
      v8f re = RR - II,   im = RI + IR;
      v8f qe = sRR - sII, qm = sRI + sIR;
      const int p = p0 + wave;
      #pragma unroll
      for (int r = 0; r < 8; ++r) {
        const int m = r + 8*hi;
        const size_t off = ((size_t)(m*COUT + o0 + mn))*NP + p;
        out1 [off] = make_float2(re[r], im[r]);
        skipb[off] = make_float2(qe[r], qm[r]);
      }
    }
  }
}

// =====================================================================
// Kernel 2/4: instance-norm stats per (b,o) channel over 4096 pixels.
// stats = {mean_re, mean_im, 1/sqrt(var+eps), 0};  var = E|z|^2 - |mean|^2
// =====================================================================
__global__ __launch_bounds__(256) void stats_kernel(const float2* __restrict__ buf,
                                                    float4* __restrict__ stats)
{
  __shared__ float sr[256], si[256], sq[256];
  const int ch = blockIdx.x, tid = threadIdx.x;
  const float2* row = buf + (size_t)ch * NP;
  float a_r = 0.f, a_i = 0.f, a_q = 0.f;
  for (int p = tid; p < NP; p += 256) {
    float2 z = row[p];
    a_r += z.x; a_i += z.y; a_q += z.x*z.x + z.y*z.y;
  }
  sr[tid] = a_r; si[tid] = a_i; sq[tid] = a_q;
  __syncthreads();
  for (int s = 128; s > 0; s >>= 1) {
    if (tid < s) { sr[tid] += sr[tid+s]; si[tid] += si[tid+s]; sq[tid] += sq[tid+s]; }
    __syncthreads();
  }
  if (tid == 0) {
    float mr = sr[0] / NP, mi = si[0] / NP;
    float var = sq[0] / NP - (mr*mr + mi*mi);
    stats[ch] = make_float4(mr, mi, rsqrtf(var + EPSN), 0.f);
  }
}

// =====================================================================
// Kernel 3: norm1 + gamma/beta + log-magnitude + elementwise *f2 (in place)
// =====================================================================
__global__ __launch_bounds__(256) void mid_kernel(float2* __restrict__ buf,
    const float4* __restrict__ st1,
    const float2* __restrict__ g1, const float2* __restrict__ b1,
    const float2* __restrict__ f2)
{
  const int ch = blockIdx.x, o = ch & (COUT-1);
  const float4 s  = st1[ch];
  const float2 g  = g1[o], be = b1[o];
  float2* row = buf + (size_t)ch * NP;
  const float2* frow = f2 + (size_t)o * NP;
  for (int p = threadIdx.x; p < NP; p += 256) {
    float2 z = row[p];
    z.x = (z.x - s.x) * s.z;  z.y = (z.y - s.y) * s.z;
    z = cadd(cmul(g, z), be);
    z = logmag(z);
    row[p] = cmul(z, frow[p]);
  }
}

// =====================================================================
// Kernel 5: norm2(out2) + norm_s(skip) -> add -> log-magnitude -> output
// =====================================================================
__global__ __launch_bounds__(256) void final_kernel(
    const float2* __restrict__ buf, const float2* __restrict__ skipb,
    const float4* __restrict__ st2, const float4* __restrict__ sts,
    const float2* __restrict__ g2, const float2* __restrict__ b2,
    const float2* __restrict__ gs, const float2* __restrict__ bs,
    float2* __restrict__ out)
{
  const int ch = blockIdx.x, o = ch & (COUT-1);
  const float4 s2 = st2[ch], ss = sts[ch];
  const float2 G2 = g2[o], B2 = b2[o], Gs = gs[o], Bs = bs[o];
  const size_t base = (size_t)ch * NP;
  for (int p = threadIdx.x; p < NP; p += 256) {
    float2 z = buf[base + p];
    z.x = (z.x - s2.x) * s2.z;  z.y = (z.y - s2.y) * s2.z;
    z = cadd(cmul(G2, z), B2);
    float2 w = skipb[base + p];
    w.x = (w.x - ss.x) * ss.z;  w.y = (w.y - ss.y) * ss.z;
    w = cadd(cmul(Gs, w), Bs);
    out[base + p] = logmag(cadd(z, w));
  }
}

extern "C" void kernel_launch(void* const* d_in, const int* in_sizes, int n_in,
                              void* d_out, int out_size, void* d_ws, size_t ws_size,
                              hipStream_t stream)
{
  (void)in_sizes; (void)n_in; (void)out_size; (void)ws_size;
  const float*  xr = (const float*) d_in[0];
  const float*  xi = (const float*) d_in[1];
  const float2* f1 = (const float2*)d_in[2];
  const float2* g1 = (const float2*)d_in[3];
  const float2* b1 = (const float2*)d_in[4];
  const float2* f2 = (const float2*)d_in[5];
  const float2* g2 = (const float2*)d_in[6];
  const float2* b2 = (const float2*)d_in[7];
  const float2* fs = (const float2*)d_in[8];
  const float2* gs = (const float2*)d_in[9];
  const float2* bs = (const float2*)d_in[10];

  float2* out1  = (float2*)d_ws;                      // 64 MB  [b][o][p] complex
  float2* skipb = out1 + (size_t)NCH * NP;            // 64 MB
  float4* st1   = (float4*)(skipb + (size_t)NCH * NP);
  float4* st2   = st1 + NCH;
  float4* sts   = st2 + NCH;

  conv_gemm_kernel<<<dim3(NP/TP), dim3(32*TP), 0, stream>>>(xr, xi, f1, fs, out1, skipb);
  stats_kernel   <<<dim3(NCH),    dim3(256),   0, stream>>>(out1,  st1);
  stats_kernel   <<<dim3(NCH),    dim3(256),   0, stream>>>(skipb, sts);
  mid_kernel     <<<dim3(NCH),    dim3(256),   0, stream>>>(out1, st1, g1, b1, f2);
  stats_kernel   <<<dim3(NCH),    dim3(256),   0, stream>>>(out1,  st2);
  final_kernel   <<<dim3(NCH),    dim3(256),   0, stream>>>(out1, skipb, st2, sts,
                                                            g2, b2, gs, bs, (float2*)d_out);
}